// RecurrentNN_27307402068532
// MI455X (gfx1250) — compile-verified
//
#include <hip/hip_runtime.h>

// LSTM (H=4, in=1) over B=4096 rows, T=2048 steps, + linear head.
// One wave32 owns 16 batch rows; recurrent matvec gates = W_hh @ h^T is a
// single V_WMMA_F32_16X16X4_F32 per timestep:
//   A (16x4)  = W_hh              (VGPR0: K=0|K=2 per half, VGPR1: K=1|K=3)
//   B (4x16)  = h^T of 16 rows    (VGPR0: K=0|K=2 rows striped on lanes)
//   C (16x16) = x*W_ih + b        (VGPR v: gate v lanes0-15, gate v+8 lanes16-31)
// D layout pairs i|g (VGPRs 0-3) and f|o (VGPRs 4-7) across wave halves.
// Cross-half exchange: v_permlane16_swap_b32 (pure VALU, confirmed on gfx1250).
// x pipelining: unconditional clamped distance-2 b128 loads (keeps loadcnt
// deterministic -> precise s_wait_loadcnt, no join-normalization stall) plus a
// counter-free global_prefetch_b8 one cacheline (~32 steps) ahead for HBM.

typedef float v2f __attribute__((ext_vector_type(2)));
typedef float v8f __attribute__((ext_vector_type(8)));

#define B_ROWS 4096
#define T_LEN  2048

#if __has_builtin(__builtin_amdgcn_permlane16_swap)
#define HAVE_PLSWAP 1
#endif

__device__ __forceinline__ float swap16(float v) {
  // ds_swizzle group-of-32: xor=0x10, or=0, and=0x1f  -> lanes l <-> l^16
  return __int_as_float(__builtin_amdgcn_ds_swizzle(__float_as_int(v), 0x401f));
}

// From a register holding {A (lanes 0-15) | B (lanes 16-31)}, produce
// loAll = A in all 32 lanes, hiAll = B in all 32 lanes.
__device__ __forceinline__ void bcast_halves(float v, bool hiLane,
                                             float& loAll, float& hiAll) {
#ifdef HAVE_PLSWAP
  (void)hiLane;
  // v_permlane16_swap_b32 d,s: d.hi <-> s.lo.  With d=s=v:
  //   result0 = {v.lo | v.lo},  result1 = {v.hi | v.hi}
  auto r = __builtin_amdgcn_permlane16_swap(__float_as_uint(v),
                                            __float_as_uint(v), false, false);
  loAll = __uint_as_float(r[0]);
  hiAll = __uint_as_float(r[1]);
#else
  float s = swap16(v);
  loAll = hiLane ? s : v;
  hiAll = hiLane ? v : s;
#endif
}

__device__ __forceinline__ float fast_tanh(float x) {
#if __has_builtin(__builtin_amdgcn_tanhf)
  return __builtin_amdgcn_tanhf(x);            // v_tanh_f32 (gfx1250 trans op)
#else
  float e = __builtin_amdgcn_exp2f(x * 2.8853900817779268f);   // 2*log2(e)
  return 1.0f - 2.0f * __builtin_amdgcn_rcpf(e + 1.0f);
#endif
}

__global__ __launch_bounds__(256, 1)
void lstm16_wmma(const float* __restrict__ x,
                 const float* __restrict__ W_ih,
                 const float* __restrict__ W_hh,
                 const float* __restrict__ b_ih,
                 const float* __restrict__ b_hh,
                 const float* __restrict__ W_lin,
                 const float* __restrict__ b_lin,
                 float* __restrict__ out) {
  const int lane   = threadIdx.x & 31;
  const int wave   = blockIdx.x * (blockDim.x >> 5) + (threadIdx.x >> 5);
  const int r      = lane & 15;          // row within 16-row tile / gate index
  const bool hi    = lane >= 16;
  const int row    = wave * 16 + r;      // batch row this lane-pair owns

  // ---- A matrix: W_hh is [16 gates x 4 hidden] row-major ----
  v2f Amat;
  Amat.x = W_hh[r * 4 + (hi ? 2 : 0)];   // VGPR0: K=0 (lo) / K=2 (hi)
  Amat.y = W_hh[r * 4 + (hi ? 3 : 1)];   // VGPR1: K=1 (lo) / K=3 (hi)

  // ---- per-lane constants for C build: gate m = v (lo half) / v+8 (hi half)
  float wsel[8], bsel[8];
#pragma unroll
  for (int v = 0; v < 8; ++v) {
    int m = v + (hi ? 8 : 0);
    wsel[v] = W_ih[m];
    bsel[v] = b_ih[m] + b_hh[m];
  }

  // activation coeffs for D VGPRs 0-3: lo half sigmoid (0.5*tanh(0.5x)+0.5),
  // hi half plain tanh.  VGPRs 4-7 are sigmoid in both halves.
  const float s1 = hi ? 1.0f : 0.5f;
  const float s2 = s1;
  const float s3 = hi ? 0.0f : 0.5f;

  float wl[4];
#pragma unroll
  for (int v = 0; v < 4; ++v) wl[v] = W_lin[v];
  const float bl = b_lin[0];

  // ---- recurrent state (identical in both wave halves) ----
  v2f Bmat; Bmat.x = 0.0f; Bmat.y = 0.0f;     // h0 = 0
  float cst[4] = {0.0f, 0.0f, 0.0f, 0.0f};    // c0 = 0

  const float* xrow = x + (size_t)row * T_LEN;
  float*       yrow = out + (size_t)row * T_LEN;

  // distance-2 prefetch: two b128 loads in flight at all times
  float4 q0 = *reinterpret_cast<const float4*>(xrow + 0);
  float4 q1 = *reinterpret_cast<const float4*>(xrow + 4);

  for (int t4 = 0; t4 < T_LEN; t4 += 4) {
    // long-range, counter-free cacheline prefetch (speculative: OOB is safe)
    __builtin_prefetch(xrow + t4 + 32, 0, 3);

    // unconditional architectural prefetch with uniform clamped offset:
    // no control-flow join -> loadcnt stays deterministic, waits stay precise
    int offn = t4 + 8;
    offn = (offn < T_LEN) ? offn : 0;
    const float4 q2 = *reinterpret_cast<const float4*>(xrow + offn);

    const float xq[4] = {q0.x, q0.y, q0.z, q0.w};
    float yq[4];

#pragma unroll
    for (int j = 0; j < 4; ++j) {
      const float xr = xq[j];

      // C = x*W_ih + (b_ih+b_hh) in WMMA C layout
      v8f C;
#pragma unroll
      for (int v = 0; v < 8; ++v) C[v] = fmaf(xr, wsel[v], bsel[v]);

      // gates = W_hh @ h^T + C   (16 rows x 16 gates, one instruction)
      v8f D = __builtin_amdgcn_wmma_f32_16x16x4_f32(
          false, Amat, false, Bmat, (short)0, C, false, false);

      // activations: VGPRs 0-3 = i(lo)/g(hi), VGPRs 4-7 = f(lo)/o(hi)
      float act[8];
#pragma unroll
      for (int v = 0; v < 4; ++v)
        act[v] = fmaf(fast_tanh(D[v] * s1), s2, s3);
#pragma unroll
      for (int v = 4; v < 8; ++v)
        act[v] = fmaf(fast_tanh(D[v] * 0.5f), 0.5f, 0.5f);

      // separate halves so every lane holds i,f,g,o for its row
      float iv[4], gv[4], fv[4], ov[4];
#pragma unroll
      for (int v = 0; v < 4; ++v) bcast_halves(act[v],     hi, iv[v], gv[v]);
#pragma unroll
      for (int v = 0; v < 4; ++v) bcast_halves(act[v + 4], hi, fv[v], ov[v]);

      float hq[4];
#pragma unroll
      for (int v = 0; v < 4; ++v) {
        cst[v] = fmaf(fv[v], cst[v], iv[v] * gv[v]);    // c = f*c + i*g
        hq[v]  = ov[v] * fast_tanh(cst[v]);             // h = o*tanh(c)
      }

      // next-step B matrix: VGPR0 = h0(lo)|h2(hi), VGPR1 = h1(lo)|h3(hi)
      Bmat.x = hi ? hq[2] : hq[0];
      Bmat.y = hi ? hq[3] : hq[1];

      // output head: y = h . W_lin + b_lin
      yq[j] = fmaf(hq[0], wl[0],
              fmaf(hq[1], wl[1],
              fmaf(hq[2], wl[2],
              fmaf(hq[3], wl[3], bl))));
    }

    if (!hi)  // both halves computed identical y; store once per row
      *reinterpret_cast<float4*>(yrow + t4) =
          make_float4(yq[0], yq[1], yq[2], yq[3]);

    q0 = q1;
    q1 = q2;
  }
}

extern "C" void kernel_launch(void* const* d_in, const int* in_sizes, int n_in,
                              void* d_out, int out_size, void* d_ws, size_t ws_size,
                              hipStream_t stream) {
  (void)in_sizes; (void)n_in; (void)d_ws; (void)ws_size; (void)out_size;
  const float* x     = (const float*)d_in[0];
  const float* W_ih  = (const float*)d_in[1];
  const float* W_hh  = (const float*)d_in[2];
  const float* b_ih  = (const float*)d_in[3];
  const float* b_hh  = (const float*)d_in[4];
  const float* W_lin = (const float*)d_in[5];
  const float* b_lin = (const float*)d_in[6];
  float* out = (float*)d_out;

  // 4096 rows / 16 rows-per-wave = 256 waves; 8 waves per 256-thread block.
  dim3 grid(B_ROWS / 16 / 8), block(256);
  lstm16_wmma<<<grid, block, 0, stream>>>(x, W_ih, W_hh, b_ih, b_hh,
                                          W_lin, b_lin, out);
}